// GAT_45157286150549
// MI455X (gfx1250) — compile-verified
//
#include <hip/hip_runtime.h>
#include <hip/hip_bf16.h>

// ---------------------------------------------------------------------------
// 2-layer GAT for MI455X (gfx1250).
//   GEMMs via V_WMMA_F32_16X16X4_F32 (fp32, matches reference precision;
//   GEMM is ~10 GFLOP total so the edge scatter/gather phase dominates and
//   lives in the 192MB L2). Segment softmax via uint-encoded float atomicMax.
// ---------------------------------------------------------------------------

typedef float v2f __attribute__((ext_vector_type(2)));
typedef float v8f __attribute__((ext_vector_type(8)));

#define GAT_NEG_SLOPE 0.2f

__device__ __forceinline__ unsigned GAT_fkey(float f) {
  // monotonic float -> uint mapping (order-preserving under unsigned compare)
  unsigned u = __float_as_uint(f);
  return (u & 0x80000000u) ? ~u : (u | 0x80000000u);
}
__device__ __forceinline__ float GAT_funkey(unsigned k) {
  return (k & 0x80000000u) ? __uint_as_float(k & 0x7FFFFFFFu)
                           : __uint_as_float(~k);
}
__device__ __forceinline__ float GAT_lrelu(float v) {
  return v > 0.0f ? v : GAT_NEG_SLOPE * v;
}

// out[i] = bias[i % C]  (pre-seed aggregation buffers with the bias)
__global__ void GAT_init_bias(float* __restrict__ out, const float* __restrict__ b,
                              long long total, int C) {
  long long i = (long long)blockIdx.x * blockDim.x + threadIdx.x;
  if (i >= total) return;
  out[i] = b[(int)(i % C)];
}

// ---------------------------------------------------------------------------
// Dual-output WMMA GEMM:  Hs = act(A) @ Bs  (stored),  and per-head attention
// logits aS[n,h] = sum_c Hs[n,h,c]*attS[h,c],  aD from act(A)@Bd (Hd never
// stored).  One 16x16 tile per wave, f32 16x16x4 WMMA, K stepped by 4.
// ---------------------------------------------------------------------------
__global__ void GAT_gemm_dual(const float* __restrict__ A,
                              const float* __restrict__ Bs,
                              const float* __restrict__ Bd,
                              const float* __restrict__ attS,
                              const float* __restrict__ attD,
                              float* __restrict__ Hout,
                              float* __restrict__ aS,
                              float* __restrict__ aD,
                              int M, int K, int NC, int Ch, int reluA) {
  const int wave = blockIdx.x * (blockDim.x >> 5) + (threadIdx.x >> 5);
  const int nT = NC >> 4;
  const int totalTiles = (M >> 4) * nT;
  if (wave >= totalTiles) return;               // wave-uniform: EXEC stays all-1s

  const int mBase = (wave / nT) << 4;
  const int nBase = (wave % nT) << 4;
  const int lane = threadIdx.x & 31;
  const int half = lane >> 4;                   // K-half select for A/B frags
  const int l16  = lane & 15;

  v8f cS = {};
  v8f cD = {};
  const float* arow = A + (size_t)(mBase + l16) * K;
  const int col = nBase + l16;

  for (int k0 = 0; k0 < K; k0 += 4) {
    const int ka = k0 + 2 * half;
    float a0 = arow[ka];
    float a1 = arow[ka + 1];
    if (reluA) { a0 = fmaxf(a0, 0.0f); a1 = fmaxf(a1, 0.0f); }
    v2f a;  a.x = a0; a.y = a1;

    v2f bs, bd;
    bs.x = Bs[(size_t)ka * NC + col];
    bs.y = Bs[(size_t)(ka + 1) * NC + col];
    bd.x = Bd[(size_t)ka * NC + col];
    bd.y = Bd[(size_t)(ka + 1) * NC + col];

    cS = __builtin_amdgcn_wmma_f32_16x16x4_f32(false, a, false, bs, (short)0, cS, false, false);
    cD = __builtin_amdgcn_wmma_f32_16x16x4_f32(false, a, false, bd, (short)0, cD, false, false);
  }

  // store Hs tile: VGPR r -> row (mBase + r + 8*half), col = nBase + l16
#pragma unroll
  for (int r = 0; r < 8; ++r) {
    Hout[(size_t)(mBase + r + 8 * half) * NC + col] = cS[r];
  }

  // attention logits: reduce att-weighted columns across the 16 lanes of each half
  const int head = nBase / Ch;                  // uniform per tile (16 | Ch)
  const int H = NC / Ch;
  const float avS = attS[col];                  // att is [H,Ch] flat == col
  const float avD = attD[col];
#pragma unroll
  for (int r = 0; r < 8; ++r) {
    float vs = cS[r] * avS;
    float vd = cD[r] * avD;
#pragma unroll
    for (int m = 1; m < 16; m <<= 1) {
      vs += __shfl_xor(vs, m, 32);
      vd += __shfl_xor(vd, m, 32);
    }
    if (l16 == 0) {
      const int row = mBase + r + 8 * half;
      atomicAdd(&aS[row * H + head], vs);
      atomicAdd(&aD[row * H + head], vd);
    }
  }
}

// pass 1: segment max over destinations (uint-encoded float atomicMax)
__global__ void GAT_edge_max(const int* __restrict__ src, const int* __restrict__ dst,
                             const float* __restrict__ aS, const float* __restrict__ aD,
                             unsigned* __restrict__ mkey, long long EH, int H) {
  long long i = (long long)blockIdx.x * blockDim.x + threadIdx.x;
  if (i >= EH) return;
  const int e = (int)(i / H), h = (int)(i % H);
  const int s = src[e], d = dst[e];
  const float v = GAT_lrelu(aS[s * H + h] + aD[d * H + h]);
  atomicMax(&mkey[d * H + h], GAT_fkey(v));
}

// pass 2: denom = segment_sum(exp(e - max))
__global__ void GAT_edge_expsum(const int* __restrict__ src, const int* __restrict__ dst,
                                const float* __restrict__ aS, const float* __restrict__ aD,
                                const unsigned* __restrict__ mkey, float* __restrict__ den,
                                long long EH, int H) {
  long long i = (long long)blockIdx.x * blockDim.x + threadIdx.x;
  if (i >= EH) return;
  const int e = (int)(i / H), h = (int)(i % H);
  const int s = src[e], d = dst[e];
  const float v = GAT_lrelu(aS[s * H + h] + aD[d * H + h]);
  const float ex = __expf(v - GAT_funkey(mkey[d * H + h]));
  atomicAdd(&den[d * H + h], ex);
}

// pass 3: out[dst] += alpha * h_src[src]   (float4 per thread; 64 (or 16)
// consecutive lanes cover one edge's full feature row -> coalesced gathers)
__global__ void GAT_edge_scatter(const int* __restrict__ src, const int* __restrict__ dst,
                                 const float* __restrict__ aS, const float* __restrict__ aD,
                                 const unsigned* __restrict__ mkey, const float* __restrict__ den,
                                 const float* __restrict__ Hsrc, float* __restrict__ out,
                                 long long total, int H, int C) {
  long long i = (long long)blockIdx.x * blockDim.x + threadIdx.x;
  if (i >= total) return;
  const int pe = (H * C) >> 2;                  // float4 groups per edge
  const int e = (int)(i / pe);
  const int j = (int)(i % pe);
  const int c4 = j << 2;
  const int h = c4 / C;
  const int s = src[e], d = dst[e];

  const float v = GAT_lrelu(aS[s * H + h] + aD[d * H + h]);
  const float ex = __expf(v - GAT_funkey(mkey[d * H + h]));
  const float alpha = ex / (den[d * H + h] + 1e-16f);

  const float4 hv = *(const float4*)(Hsrc + (size_t)s * (H * C) + c4);
  float* op = out + (size_t)d * (H * C) + c4;
  atomicAdd(op + 0, hv.x * alpha);
  atomicAdd(op + 1, hv.y * alpha);
  atomicAdd(op + 2, hv.z * alpha);
  atomicAdd(op + 3, hv.w * alpha);
}

// ---------------------------------------------------------------------------
extern "C" void kernel_launch(void* const* d_in, const int* in_sizes, int n_in,
                              void* d_out, int out_size, void* d_ws, size_t ws_size,
                              hipStream_t stream) {
  (void)n_in; (void)out_size; (void)ws_size;

  const float* x     = (const float*)d_in[0];
  const int*   ei    = (const int*)d_in[1];
  const float* Wsrc1 = (const float*)d_in[2];
  const float* Wdst1 = (const float*)d_in[3];
  const float* attS1 = (const float*)d_in[4];
  const float* attD1 = (const float*)d_in[5];
  const float* b1    = (const float*)d_in[6];
  const float* Wsrc2 = (const float*)d_in[7];
  const float* Wdst2 = (const float*)d_in[8];
  const float* attS2 = (const float*)d_in[9];
  const float* attD2 = (const float*)d_in[10];
  const float* b2    = (const float*)d_in[11];

  const int FIN = 128, HEADS = 8, HID = 32, EMB = 64;
  const int HC1 = HEADS * HID;                  // 256
  const long long N = in_sizes[0] / FIN;        // 50000 (divisible by 16)
  const long long E = in_sizes[1] / 2;          // 800000
  const int* src = ei;
  const int* dst = ei + E;

  // workspace carve-up (all chunk sizes are multiples of 16 bytes)
  float* p = (float*)d_ws;
  float* h1   = p; p += N * HC1;                // layer-1 h_src [N,256]
  float* h2   = p; p += N * EMB;                // layer-2 h_src [N,64]
  float* aS1  = p; p += N * HEADS;              // ---- zero-init block starts
  float* aD1  = p; p += N * HEADS;
  float* den1 = p; p += N * HEADS;
  float* aS2  = p; p += N;
  float* aD2  = p; p += N;
  float* den2 = p; p += N;
  unsigned* mk1 = (unsigned*)p; p += N * HEADS;
  unsigned* mk2 = (unsigned*)p; p += N;         // ---- zero-init block ends
  float* out1 = p; p += N * HC1;                // layer-1 aggregation [N,256]
  float* outF = (float*)d_out;                  // final [N,64]

  const size_t zbytes = (size_t)N * (3 * HEADS + 3 + HEADS + 1) * sizeof(float);
  hipMemsetAsync(aS1, 0, zbytes, stream);       // zeros aS/aD/den + mkey (key 0 < any float)

  const int TPB = 256;
  auto nb = [](long long n, int t) { return (unsigned)((n + t - 1) / t); };

  // seed aggregation buffers with bias (folds "+ b" into the scatter)
  GAT_init_bias<<<nb(N * HC1, TPB), TPB, 0, stream>>>(out1, b1, N * HC1, HC1);
  GAT_init_bias<<<nb(N * EMB, TPB), TPB, 0, stream>>>(outF, b2, N * EMB, EMB);

  // ---- layer 1 ----
  {
    const long long tiles = (N / 16) * (HC1 / 16);
    GAT_gemm_dual<<<nb(tiles * 32, TPB), TPB, 0, stream>>>(
        x, Wsrc1, Wdst1, attS1, attD1, h1, aS1, aD1,
        (int)N, FIN, HC1, HID, /*reluA=*/0);
  }
  GAT_edge_max<<<nb(E * HEADS, TPB), TPB, 0, stream>>>(src, dst, aS1, aD1, mk1, E * HEADS, HEADS);
  GAT_edge_expsum<<<nb(E * HEADS, TPB), TPB, 0, stream>>>(src, dst, aS1, aD1, mk1, den1, E * HEADS, HEADS);
  {
    const long long total = E * (long long)(HEADS * HID / 4);
    GAT_edge_scatter<<<nb(total, TPB), TPB, 0, stream>>>(
        src, dst, aS1, aD1, mk1, den1, h1, out1, total, HEADS, HID);
  }

  // ---- layer 2 (A = relu(out1), bias already seeded into out1) ----
  {
    const long long tiles = (N / 16) * (EMB / 16);
    GAT_gemm_dual<<<nb(tiles * 32, TPB), TPB, 0, stream>>>(
        out1, Wsrc2, Wdst2, attS2, attD2, h2, aS2, aD2,
        (int)N, HC1, EMB, EMB, /*reluA=*/1);
  }
  GAT_edge_max<<<nb(E, TPB), TPB, 0, stream>>>(src, dst, aS2, aD2, mk2, E, 1);
  GAT_edge_expsum<<<nb(E, TPB), TPB, 0, stream>>>(src, dst, aS2, aD2, mk2, den2, E, 1);
  {
    const long long total = E * (long long)(EMB / 4);
    GAT_edge_scatter<<<nb(total, TPB), TPB, 0, stream>>>(
        src, dst, aS2, aD2, mk2, den2, h2, outF, total, 1, EMB);
  }
}